// DynamicsEquation_33243046871050
// MI455X (gfx1250) — compile-verified
//
#include <hip/hip_runtime.h>
#include <hip/hip_bf16.h>
#include <stdint.h>

#define TILE  1024      // edges staged per block
#define BLOCK 256       // 8 waves (wave32)

#define GLOBAL_AS __attribute__((address_space(1)))
#define LDS_AS    __attribute__((address_space(3)))

typedef int v4i __attribute__((ext_vector_type(4)));

typedef GLOBAL_AS v4i* gvec_t;   // global-AS int4*, builtin param 1
typedef LDS_AS    v4i* lvec_t;   // LDS-AS    int4*, builtin param 2

__device__ __forceinline__ gvec_t to_global(const void* p) {
    return (gvec_t)(uintptr_t)p;                 // generic global addr == AS1 addr
}
__device__ __forceinline__ lvec_t to_lds(void* p) {
    return (lvec_t)(uint32_t)(uintptr_t)p;       // low 32 bits = LDS byte offset
}

__device__ __forceinline__ float fast_tanh(float x) {
#if __has_builtin(__builtin_amdgcn_tanhf)
    return __builtin_amdgcn_tanhf(x);            // v_tanh_f32 (gfx1250 TRANS op)
#elif __has_builtin(__builtin_amdgcn_tanh_f32)
    return __builtin_amdgcn_tanh_f32(x);
#else
    return tanhf(x);
#endif
}

__device__ __forceinline__ void wait_async0() {
#if __has_builtin(__builtin_amdgcn_s_wait_asynccnt)
    __builtin_amdgcn_s_wait_asynccnt(0);
#else
    asm volatile("s_wait_asynccnt 0" ::: "memory");
#endif
}

// ---------------------------------------------------------------------------
// Edge kernel: stage (row,col,w) tile into LDS with async B128 copies
// (uniform per-block branch -> no EXEC divergence on the hot path), then
// 4 lanes per edge gather float4 slices of both endpoint rows, compute
// sin(xr-xc)*w and scatter-add with no-return f32 global atomics.
// ---------------------------------------------------------------------------
__global__ __launch_bounds__(BLOCK) void edge_scatter_kernel(
    const float* __restrict__ state,
    const int*   __restrict__ row,
    const int*   __restrict__ col,
    const float* __restrict__ w,
    float*       __restrict__ agg,
    int E)
{
    __shared__ __attribute__((aligned(16))) int   rowS[TILE];
    __shared__ __attribute__((aligned(16))) int   colS[TILE];
    __shared__ __attribute__((aligned(16))) float wS[TILE];

    const int e0 = blockIdx.x * TILE;
    const int tb = threadIdx.x * 4;       // this thread stages elements [tb, tb+3]
    const bool full_tile = (e0 + TILE <= E);   // uniform across the block

#if __has_builtin(__builtin_amdgcn_global_load_async_to_lds_b128)
    if (full_tile) {
        __builtin_amdgcn_global_load_async_to_lds_b128(
            to_global(row + e0 + tb), to_lds(&rowS[tb]), 0, 0);
        __builtin_amdgcn_global_load_async_to_lds_b128(
            to_global(col + e0 + tb), to_lds(&colS[tb]), 0, 0);
        __builtin_amdgcn_global_load_async_to_lds_b128(
            to_global(w   + e0 + tb), to_lds(&wS[tb]),   0, 0);
        wait_async0();
    } else
#endif
    {
        for (int k = 0; k < 4; ++k) {
            int e = e0 + tb + k;
            if (e < E) { rowS[tb + k] = row[e]; colS[tb + k] = col[e]; wS[tb + k] = w[e]; }
        }
    }
    __syncthreads();

    const int tmax = full_tile ? TILE : (E - e0);    // hoisted loop bound
    const int sub  = threadIdx.x & 3;                // which float4 of the 16 dims
    const float4* __restrict__ state4 = (const float4*)state;

    for (int t = (int)(threadIdx.x >> 2); t < tmax; t += BLOCK / 4) {
        const int   r  = rowS[t];
        const int   c  = colS[t];
        const float wt = wS[t];

        const float4 xr = state4[(size_t)r * 4 + sub];   // 64B/edge across 4 lanes
        const float4 xc = state4[(size_t)c * 4 + sub];

        float4 v;
        v.x = __sinf(xr.x - xc.x) * wt;   // v_sin_f32 hardware transcendental
        v.y = __sinf(xr.y - xc.y) * wt;
        v.z = __sinf(xr.z - xc.z) * wt;
        v.w = __sinf(xr.w - xc.w) * wt;

        float* dst = agg + (size_t)c * 16 + sub * 4;
        atomicAdd(dst + 0, v.x);          // global_atomic_add_f32, no return
        atomicAdd(dst + 1, v.y);
        atomicAdd(dst + 2, v.z);
        atomicAdd(dst + 3, v.w);
    }
}

// ---------------------------------------------------------------------------
// Node kernel: produce all five output sections with float4 (B128) traffic.
// agg (section 3) was accumulated in place by the edge kernel.
// ---------------------------------------------------------------------------
__global__ __launch_bounds__(256) void node_finalize_kernel(
    const float* __restrict__ state,
    const float* __restrict__ agg,
    float* __restrict__ out0,
    float* __restrict__ outState,
    float* __restrict__ outSd,
    float* __restrict__ outId,
    int n4)
{
    const int i = blockIdx.x * blockDim.x + threadIdx.x;
    if (i >= n4) return;

    const float4 x = ((const float4*)state)[i];
    const float4 a = ((const float4*)agg)[i];

    float4 sd; sd.x = -x.x; sd.y = -x.y; sd.z = -x.z; sd.w = -x.w;
    float4 id; id.x = fast_tanh(a.x); id.y = fast_tanh(a.y);
               id.z = fast_tanh(a.z); id.w = fast_tanh(a.w);
    float4 o0; o0.x = sd.x + id.x; o0.y = sd.y + id.y;
               o0.z = sd.z + id.z; o0.w = sd.w + id.w;

    ((float4*)out0)[i]     = o0;
    ((float4*)outState)[i] = x;
    ((float4*)outSd)[i]    = sd;
    ((float4*)outId)[i]    = id;
}

extern "C" void kernel_launch(void* const* d_in, const int* in_sizes, int n_in,
                              void* d_out, int out_size, void* d_ws, size_t ws_size,
                              hipStream_t stream) {
    const float* state = (const float*)d_in[0];
    const int*   row   = (const int*)d_in[1];
    const int*   col   = (const int*)d_in[2];
    const float* w     = (const float*)d_in[3];

    const int N16 = in_sizes[0];   // N * DIM  (1,600,000)
    const int E   = in_sizes[1];   // 3,200,000

    float* out      = (float*)d_out;
    float* out0     = out;                      // self_diff + interact_diff
    float* outState = out + (size_t)1 * N16;    // state_input copy
    float* outSd    = out + (size_t)2 * N16;    // self_diff
    float* agg      = out + (size_t)3 * N16;    // aggregated_sum (accumulated here)
    float* outId    = out + (size_t)4 * N16;    // interact_diff

    // zero the accumulator slice each call (graph-capture-safe memset node)
    (void)hipMemsetAsync(agg, 0, (size_t)N16 * sizeof(float), stream);

    const int nblk = (E + TILE - 1) / TILE;
    edge_scatter_kernel<<<nblk, BLOCK, 0, stream>>>(state, row, col, w, agg, E);

    const int n4 = N16 / 4;
    node_finalize_kernel<<<(n4 + 255) / 256, 256, 0, stream>>>(
        state, agg, out0, outState, outSd, outId, n4);
}